// SourceModuleHnNSF_274877907372
// MI455X (gfx1250) — compile-verified
//
#include <hip/hip_runtime.h>
#include <stdint.h>

// ---------------- problem constants (from reference) ----------------
#define B_    8
#define T_    2000
#define UPP_  480
#define S_    (T_ * UPP_)          // 960000
#define NH    9                    // HARMONIC_NUM + 1
#define SR_   48000
#define SINE_AMP_   0.1f
#define NOISE_STD_  0.003f

#define KF    10                   // frames per block in main kernel
#define TPAD  2048                 // padded frame count for ws layout

typedef unsigned int v4u __attribute__((ext_vector_type(4)));
typedef int          v8i __attribute__((ext_vector_type(8)));
typedef int          v4i __attribute__((ext_vector_type(4)));

// ws (float) layout:
//   [0,          8*TPAD)          radRev   [b][t]  = f0/SR (revolutions/sample)
//   [8*TPAD,    16*TPAD)          baseFrac [b][t]  = fract(exclusive frame prefix, rev)
//   [16*TPAD,   16*TPAD+128)      Cw [b][h] = w[h]*cos(2*pi*rand_ini)
//   [16*TPAD+128, 16*TPAD+256)    Sw [b][h] = w[h]*sin(2*pi*rand_ini)

// ------------------------------------------------------------------
// Kernel 1: per-batch-row exclusive prefix over 2000 frames (double),
// reduced mod 1 revolution, plus per-(b,h) phase/weight constants.
// ------------------------------------------------------------------
__global__ __launch_bounds__(256)
void prep_kernel(const float* __restrict__ f0,
                 const float* __restrict__ rand_ini,
                 const float* __restrict__ w,
                 float* __restrict__ ws)
{
    __shared__ double sc[256];
    const int b   = blockIdx.x;
    const int tid = threadIdx.x;
    const int t0  = tid * 8;

    double loc[8];
    double run = 0.0;
    #pragma unroll
    for (int i = 0; i < 8; ++i) {
        int t = t0 + i;
        float f = (t < T_) ? f0[b * T_ + t] : 0.0f;
        loc[i] = run;                                   // chunk-exclusive prefix
        run += (double)f * ((double)UPP_ / (double)SR_);// revolutions per frame
        if (t < T_) ws[b * TPAD + t] = f * (1.0f / (float)SR_);
    }

    // Hillis-Steele inclusive scan of per-thread chunk sums (doubles in LDS)
    sc[tid] = run;
    __syncthreads();
    for (int off = 1; off < 256; off <<= 1) {
        double v = (tid >= off) ? sc[tid - off] : 0.0;
        __syncthreads();
        sc[tid] += v;
        __syncthreads();
    }
    double offset = sc[tid] - run;                      // exclusive across chunks

    #pragma unroll
    for (int i = 0; i < 8; ++i) {
        int t = t0 + i;
        if (t < T_) {
            double p  = offset + loc[i];
            double fr = p - floor(p);                   // mod 1 revolution
            ws[8 * TPAD + b * TPAD + t] = (float)fr;
        }
    }

    if (tid < NH) {
        float r   = rand_ini[b * NH + tid];             // already in revolutions
        float wt  = w[tid];
        float ang = r * 6.28318530717958647692f;
        ws[16 * TPAD +       b * 16 + tid] = wt * cosf(ang);
        ws[16 * TPAD + 128 + b * 16 + tid] = wt * sinf(ang);
    }
}

// ------------------------------------------------------------------
// Kernel 2: 480 threads/block (15 wave32), KF frames per block.
// Noise tile DMA'd into LDS via the Tensor Data Mover; harmonics via
// one v_sin/v_cos + angle-addition recurrence.
// ------------------------------------------------------------------
__global__ __launch_bounds__(UPP_)
void src_kernel(const float* __restrict__ noise_raw,
                const float* __restrict__ bvec,
                const float* __restrict__ ws,
                float* __restrict__ out)
{
    __shared__ float lds_noise[KF * UPP_];              // 4800 floats = 19200 B

    const int b  = blockIdx.y;
    const int t0 = blockIdx.x * KF;
    const int j  = threadIdx.x;

    // ---- TDM: async-copy this block's noise tile (4800 B32) into LDS ----
    if (j < 32) {                                       // wave 0 only (EXEC ignored by TDM)
        uint64_t gaddr = (uint64_t)(uintptr_t)(noise_raw + (size_t)b * S_ + (size_t)t0 * UPP_);
        uint32_t laddr = (uint32_t)(uintptr_t)(&lds_noise[0]); // low 32 bits = LDS offset
        const uint32_t NE = (uint32_t)(KF * UPP_);      // 4800 elements, 1-D tile

        v4u g0;
        g0.x = 1u;                                      // count=1, user mode
        g0.y = laddr;                                   // lds_addr (bytes)
        g0.z = (uint32_t)gaddr;                         // global_addr[31:0]
        g0.w = (uint32_t)(gaddr >> 32) | (2u << 30);    // global_addr[56:32] | type=2

        v8i g1;
        g1[0] = (int)(2u << 16);                        // wg_mask=0, data_size=4B
        g1[1] = (int)((NE & 0xFFFFu) << 16);            // tensor_dim0[15:0]
        g1[2] = (int)((NE >> 16) | (1u << 16));         // tensor_dim0[31:16] | tensor_dim1=1
        g1[3] = (int)((NE & 0xFFFFu) << 16);            // tile_dim0 = 4800
        g1[4] = 0;                                      // tile_dim1/2 unused (1-D)
        g1[5] = (int)NE;                                // tensor_dim0_stride
        g1[6] = 0;
        g1[7] = 0;

        v4i g2 = {0, 0, 0, 0};
        v4i g3 = {0, 0, 0, 0};
        v8i g4 = {0, 0, 0, 0, 0, 0, 0, 0};              // extra group (unused, 1-D tile)
        __builtin_amdgcn_tensor_load_to_lds(g0, g1, g2, g3, g4, 0);
        __builtin_amdgcn_s_wait_tensorcnt(0);
    }
    __syncthreads();

    const float bb = bvec[0];
    float Cw[NH], Sw[NH];
    #pragma unroll
    for (int h = 0; h < NH; ++h) {
        Cw[h] = ws[16 * TPAD +       b * 16 + h];
        Sw[h] = ws[16 * TPAD + 128 + b * 16 + h];
    }
    const float harUnvoiced = tanhf(bb) * SINE_AMP_;    // sines==0 when uv==0

    #pragma unroll
    for (int f = 0; f < KF; ++f) {
        const int    t    = t0 + f;
        const float  rad  = ws[b * TPAD + t];           // block-uniform -> SMEM
        const float  bphs = ws[8 * TPAD + b * TPAD + t];
        const size_t o    = (size_t)b * S_ + (size_t)t * UPP_ + j;

        float har, uv;
        if (rad > 0.0f) {                               // frame-uniform branch
            uv = 1.0f;
            float p = fmaf((float)(j + 1), rad, bphs);  // revolutions
            p = p - floorf(p);                          // [0,1): native v_sin domain
            float sA = __builtin_amdgcn_sinf(p);        // sin(2*pi*p)
            float cA = __builtin_amdgcn_cosf(p);
            float sh = sA, ch = cA;                     // h = 1
            float acc = bb;
            #pragma unroll
            for (int h = 0; h < NH; ++h) {
                acc = fmaf(sh, Cw[h], acc);             // w_h*sin(h*th + phi_h)
                acc = fmaf(ch, Sw[h], acc);
                float sn = fmaf(sh, cA,  ch * sA);      // angle addition -> h+1
                float cn = fmaf(ch, cA, -sh * sA);
                sh = sn; ch = cn;
            }
            har = tanhf(acc) * SINE_AMP_;
        } else {
            uv  = 0.0f;
            har = harUnvoiced;
        }

        float nz = lds_noise[f * UPP_ + j] * NOISE_STD_;

        out[o]                         = har;           // har   [B,1,S]
        out[(size_t)B_ * S_ + o]       = nz;            // noise [B,1,S]
        out[2 * (size_t)B_ * S_ + o]   = uv;            // uv    [B,1,S]
    }
}

// ------------------------------------------------------------------
extern "C" void kernel_launch(void* const* d_in, const int* in_sizes, int n_in,
                              void* d_out, int out_size, void* d_ws, size_t ws_size,
                              hipStream_t stream)
{
    (void)in_sizes; (void)n_in; (void)out_size; (void)ws_size;
    const float* f0        = (const float*)d_in[0];
    const float* rand_ini  = (const float*)d_in[1];
    const float* noise_raw = (const float*)d_in[2];
    const float* w         = (const float*)d_in[3];
    const float* bvec      = (const float*)d_in[4];
    float* ws  = (float*)d_ws;
    float* out = (float*)d_out;

    prep_kernel<<<dim3(B_), dim3(256), 0, stream>>>(f0, rand_ini, w, ws);
    src_kernel<<<dim3(T_ / KF, B_), dim3(UPP_), 0, stream>>>(noise_raw, bvec, ws, out);
}